// GraphEdgeWiseAttention_39127152066908
// MI455X (gfx1250) — compile-verified
//
#include <hip/hip_runtime.h>
#include <hip/hip_bf16.h>
#include <math.h>

#define N_NODES 50000
#define N_EDGES 800000
#define FEAT    128

typedef __attribute__((ext_vector_type(2))) float v2f;
typedef __attribute__((ext_vector_type(8))) float v8f;

// ---------------------------------------------------------------------------
// Kernel 1: zero the per-node accumulator (N_NODES floats) + 1 total scalar.
// ---------------------------------------------------------------------------
__global__ void gea_zero_kernel(float* __restrict__ p, int n) {
    int i = blockIdx.x * blockDim.x + threadIdx.x;
    if (i < n) p[i] = 0.0f;
}

// ---------------------------------------------------------------------------
// Kernel 2: edge scatter. Only feature column 0 of the SpMM output feeds the
// final scalar, so we gather exactly one float per edge:
//   acc[row[e]] += w[e] * feat[col[e] * FEAT + 0]
// w/row/col loads are fully coalesced; the single-float gather hits random
// cachelines of a 25.6MB array that is resident in the 192MB L2. We prefetch
// the next iteration's gather target (emits global_prefetch_b8 on gfx1250).
// ---------------------------------------------------------------------------
__global__ void gea_edge_scatter_kernel(const float* __restrict__ feat,
                                        const float* __restrict__ w,
                                        const int*   __restrict__ row,
                                        const int*   __restrict__ col,
                                        float*       __restrict__ acc) {
    const int stride = gridDim.x * blockDim.x;
    for (int e = blockIdx.x * blockDim.x + threadIdx.x; e < N_EDGES; e += stride) {
        int nxt = e + stride;
        if (nxt < N_EDGES) {
            // Speculative prefetch of next gather target into cache hierarchy.
            __builtin_prefetch(&feat[(size_t)col[nxt] * FEAT], 0, 0);
        }
        float msg = w[e] * feat[(size_t)col[e] * FEAT];
        atomicAdd(&acc[row[e]], msg);
    }
}

// ---------------------------------------------------------------------------
// Kernel 3: total += sum_i relu(acc[i]), with the 32-lane wave reduction done
// by V_WMMA_F32_16X16X4_F32 (matmul-by-ones trick, exact in f32):
//   A(16x4): lane m (m<16) supplies A[m,0]=p_m (a.x) and A[m,1]=0 (a.y);
//            lane 16+m supplies A[m,2]=p_{16+m}, A[m,3]=0.
//   B(4x16) = all ones  ->  D[m,n] = p_m + p_{16+m}  (independent of n)
//   D layout: VGPR j, lanes 0..15 hold D[j,n]=s_j ; lanes 16..31 hold s_{8+j}
// So sum(c[0..7]) gives S_lo on lanes 0..15 and S_hi on lanes 16..31; one
// shfl_xor(16) completes the full 32-lane sum on every lane.
// All 8 waves of the 256-thread block reach the WMMA with full EXEC.
// ---------------------------------------------------------------------------
__global__ void gea_relu_reduce_kernel(const float* __restrict__ acc,
                                       float*       __restrict__ total) {
    const int stride = gridDim.x * blockDim.x;
    float p = 0.0f;
    for (int i = blockIdx.x * blockDim.x + threadIdx.x; i < N_NODES; i += stride)
        p += fmaxf(acc[i], 0.0f);

    v2f a; a.x = p;    a.y = 0.0f;   // A-matrix: partial in K=0 slot
    v2f b; b.x = 1.0f; b.y = 1.0f;   // B-matrix: all ones
    v8f c = {};
    // 8 args: (neg_a, A, neg_b, B, c_mod, C, reuse_a, reuse_b)
    c = __builtin_amdgcn_wmma_f32_16x16x4_f32(
        /*neg_a=*/false, a, /*neg_b=*/false, b,
        /*c_mod=*/(short)0, c, /*reuse_a=*/false, /*reuse_b=*/false);

    float s = c[0] + c[1] + c[2] + c[3] + c[4] + c[5] + c[6] + c[7];
    s += __shfl_xor(s, 16, 32);      // combine lower/upper half-wave sums

    if ((threadIdx.x & 31) == 0)
        atomicAdd(total, s);         // one atomic per wave
}

// ---------------------------------------------------------------------------
// Kernel 4: final sigmoid of the accumulated scalar.
// ---------------------------------------------------------------------------
__global__ void gea_sigmoid_kernel(const float* __restrict__ total,
                                   float*       __restrict__ out) {
    if (threadIdx.x == 0 && blockIdx.x == 0)
        out[0] = 1.0f / (1.0f + expf(-total[0]));
}

// ---------------------------------------------------------------------------
extern "C" void kernel_launch(void* const* d_in, const int* in_sizes, int n_in,
                              void* d_out, int out_size, void* d_ws, size_t ws_size,
                              hipStream_t stream) {
    const float* feat = (const float*)d_in[0];  // [N_NODES, FEAT] f32
    const float* w    = (const float*)d_in[1];  // [N_EDGES] f32
    const int*   row  = (const int*)  d_in[2];  // [N_EDGES] i32
    const int*   col  = (const int*)  d_in[3];  // [N_EDGES] i32

    float* acc   = (float*)d_ws;                // [N_NODES] per-node accumulator
    float* total = acc + N_NODES;               // [1] scalar accumulator
    float* out   = (float*)d_out;

    const int nz = N_NODES + 1;
    gea_zero_kernel<<<(nz + 255) / 256, 256, 0, stream>>>(acc, nz);

    // 512 blocks x 256 threads: ~6 edges/thread grid-stride, keeps all WGPs
    // busy while staying latency-tolerant for the random gathers + atomics.
    gea_edge_scatter_kernel<<<512, 256, 0, stream>>>(feat, w, row, col, acc);

    gea_relu_reduce_kernel<<<64, 256, 0, stream>>>(acc, total);

    gea_sigmoid_kernel<<<1, 32, 0, stream>>>(total, out);
}